// RNNFFT_78941498901164
// MI455X (gfx1250) — compile-verified
//
#include <hip/hip_runtime.h>

typedef __attribute__((ext_vector_type(2))) float v2f;
typedef __attribute__((ext_vector_type(8))) float v8f;

#define WPB 8            // waves per block (256 threads, wave32)
#define NCOLS 1024       // row length N

// ---------------------------------------------------------------------------
// Kernel 1: compose butterfly levels 9..5 (all inside aligned 32-blocks, all
// with residual) into one 32x32 matrix B.  Column j = levels 9..5 applied to
// basis vector e_j.  Tiny one-shot kernel, 32 threads.
// ---------------------------------------------------------------------------
__global__ void build_bmat_kernel(const float* __restrict__ bw,
                                  const float* __restrict__ lw,
                                  float* __restrict__ Bmat) {
  constexpr int OFF[10] = {0, 1024, 1536, 1792, 1920, 1984, 2016, 2032, 2040, 2044};
  const int j = threadIdx.x;  // basis column 0..31
  float e[32], z[32];
#pragma unroll
  for (int t = 0; t < 32; ++t) { e[t] = (t == j) ? 1.0f : 0.0f; z[t] = e[t]; }
#pragma unroll
  for (int l = 9; l >= 5; --l) {
    const int n = 1024 >> l, h = n >> 1, O = OFF[l];
    const float c00 = lw[l*4+0], c01 = lw[l*4+1], c10 = lw[l*4+2], c11 = lw[l*4+3];
    float nz[32];
#pragma unroll
    for (int base = 0; base < 32; base += n) {
#pragma unroll
      for (int k = 0; k < h; ++k) {
        const float u0 = bw[O + k]     * z[base + k];
        const float u1 = bw[O + h + k] * z[base + h + k];
        nz[base + k]     = e[base + k]     + c00 * u0 + c10 * u1;   // RES true for l>=2
        nz[base + h + k] = e[base + h + k] + c01 * u0 + c11 * u1;
      }
    }
#pragma unroll
    for (int t = 0; t < 32; ++t) z[t] = nz[t];
  }
#pragma unroll
  for (int t = 0; t < 32; ++t) Bmat[t * 32 + j] = z[t];  // row t, column j
}

// ---------------------------------------------------------------------------
// Kernel 2: main pass.  One wave per row (32 lanes x 32 f32 = 1024).
//   Stage 1: z1[j,B] = sum_{j'} Bmat[j][j'] * x[32B+j']  via fp32 WMMA 16x16x4
//   Stage 2: levels 4..0 cross-block butterflies via shfl_xor (lane = block B)
// ---------------------------------------------------------------------------
__global__ __launch_bounds__(256)
void fft_main_kernel(const float* __restrict__ x, const float* __restrict__ bw,
                     const float* __restrict__ lw, const float* __restrict__ Bmat,
                     float* __restrict__ out, int rows, int rpw) {
  constexpr int OFF5[5] = {0, 1024, 1536, 1792, 1920};  // bw offsets, levels 0..4
  __shared__ alignas(16) float sBW[2048];      // staged butterfly weights
  __shared__ alignas(16) float sB[1024];       // composed 32x32 matrix
  __shared__ alignas(16) float sX[WPB][NCOLS]; // per-wave operand / remap area

  const int tid = threadIdx.x, lane = tid & 31, wave = tid >> 5;

  for (int i = tid; i < 2046; i += 256) sBW[i] = bw[i];
  for (int i = tid; i < 1024; i += 256) sB[i] = Bmat[i];
  __syncthreads();

  // A-fragments of B (16x4 f32 layout: lanes0-15 hold M, VGPR0=K0/K2, VGPR1=K1/K3)
  const int mrow = lane & 15, khalf = (lane >> 4) * 2;
  v2f Af[2][8];
#pragma unroll
  for (int jt = 0; jt < 2; ++jt)
#pragma unroll
    for (int ks = 0; ks < 8; ++ks) {
      const float* p = &sB[(jt * 16 + mrow) * 32 + ks * 4 + khalf];
      v2f a; a.x = p[0]; a.y = p[1];
      Af[jt][ks] = a;
    }

  const int gw = blockIdx.x * WPB + wave;
  float* const sx = sX[wave];

  for (int it = 0; it < rpw; ++it) {
    int row = gw * rpw + it;
    row = (row < rows) ? row : rows - 1;  // clamp: duplicate writes are identical
    const float4* xr = (const float4*)(x + (size_t)row * NCOLS);

    // load row: lane owns 32 contiguous elements (block B = lane)
    float xs[32];
#pragma unroll
    for (int q = 0; q < 8; ++q) {
      float4 v = xr[lane * 8 + q];
      xs[4*q] = v.x; xs[4*q+1] = v.y; xs[4*q+2] = v.z; xs[4*q+3] = v.w;
      ((float4*)sx)[lane * 8 + q] = v;
    }
    if (it + 1 < rpw) __builtin_prefetch(x + ((size_t)row + 1) * NCOLS + lane * 32, 0, 0);
    __syncthreads();

    // B-fragments: X[j'=K, B=N] (4x16 f32: lanes0-15 hold N, VGPR0=K0/K2, VGPR1=K1/K3)
    v2f Bf[2][8];
    const int ncol = lane & 15;
#pragma unroll
    for (int bt = 0; bt < 2; ++bt)
#pragma unroll
      for (int ks = 0; ks < 8; ++ks) {
        const float* p = &sx[(bt * 16 + ncol) * 32 + ks * 4 + khalf];
        v2f b; b.x = p[0]; b.y = p[1];
        Bf[bt][ks] = b;
      }

    // z1 = B * Xblk : 2x2 tiles of 16x16, K=32 in 8 steps of 4
    v8f acc[2][2];
#pragma unroll
    for (int jt = 0; jt < 2; ++jt)
#pragma unroll
      for (int bt = 0; bt < 2; ++bt) {
        v8f c = {};
#pragma unroll
        for (int ks = 0; ks < 8; ++ks)
          c = __builtin_amdgcn_wmma_f32_16x16x4_f32(false, Af[jt][ks],
                                                    false, Bf[bt][ks],
                                                    (short)0, c, false, false);
        acc[jt][bt] = c;
      }
    __syncthreads();

    // scatter D tiles (lanes hold N=B, VGPR v holds M=j / j+8) into [B][j] layout
    const int Brow = lane & 15, jadd = (lane >> 4) * 8;
#pragma unroll
    for (int jt = 0; jt < 2; ++jt)
#pragma unroll
      for (int bt = 0; bt < 2; ++bt) {
        float* p = &sx[(bt * 16 + Brow) * 32 + jt * 16 + jadd];
        v8f c = acc[jt][bt];
        ((float4*)p)[0] = make_float4(c[0], c[1], c[2], c[3]);
        ((float4*)p)[1] = make_float4(c[4], c[5], c[6], c[7]);
      }
    __syncthreads();

    // gather: lane = block B holds z[j], j = 0..31
    float z[32];
#pragma unroll
    for (int q = 0; q < 8; ++q) {
      float4 v = ((const float4*)(sx + lane * 32))[q];
      z[4*q] = v.x; z[4*q+1] = v.y; z[4*q+2] = v.z; z[4*q+3] = v.w;
    }

    // levels 4..0: cross-block butterflies, block stride m = 16>>l (lane-xor mask)
#pragma unroll
    for (int l = 4; l >= 0; --l) {
      const int m = 16 >> l;
      const int i = (lane >> (4 - l)) & 1;  // low/high half of the level-l block
      const float c00 = lw[l*4+0], c01 = lw[l*4+1], c10 = lw[l*4+2], c11 = lw[l*4+3];
      const float cA = i ? c11 : c00;
      const float cB = i ? c01 : c10;
      const int wb = OFF5[l] + 32 * (lane & (2 * m - 1));  // bw[OFF + (t mod n_l)]
      float w[32];
#pragma unroll
      for (int q = 0; q < 8; ++q) {
        float4 v = ((const float4*)(sBW + wb))[q];
        w[4*q] = v.x; w[4*q+1] = v.y; w[4*q+2] = v.z; w[4*q+3] = v.w;
      }
#pragma unroll
      for (int jj = 0; jj < 32; ++jj) {
        const float u  = w[jj] * z[jj];
        const float up = __shfl_xor(u, m, 32);
        const float res = (l == 1) ? 0.0f : xs[jj];   // RES[1] is the only False
        z[jj] = fmaf(cA, u, fmaf(cB, up, res));
      }
    }

    float4* orow = (float4*)(out + (size_t)row * NCOLS);
#pragma unroll
    for (int q = 0; q < 8; ++q)
      orow[lane * 8 + q] = make_float4(z[4*q], z[4*q+1], z[4*q+2], z[4*q+3]);
    __syncthreads();  // keep trip uniform before sx is rewritten
  }
}

// ---------------------------------------------------------------------------
extern "C" void kernel_launch(void* const* d_in, const int* in_sizes, int n_in,
                              void* d_out, int out_size, void* d_ws, size_t ws_size,
                              hipStream_t stream) {
  const float* x  = (const float*)d_in[0];
  const float* bw = (const float*)d_in[1];
  const float* lw = (const float*)d_in[2];
  float* out  = (float*)d_out;
  float* Bmat = (float*)d_ws;  // 1024 floats of scratch

  build_bmat_kernel<<<1, 32, 0, stream>>>(bw, lw, Bmat);

  const int rows = in_sizes[0] / NCOLS;  // 4*4096 = 16384
  const int rpw  = 4;                    // rows per wave
  const int blocks = (rows + WPB * rpw - 1) / (WPB * rpw);  // 512
  fft_main_kernel<<<blocks, 256, 0, stream>>>(x, bw, lw, Bmat, out, rows, rpw);
}